// Generator_69054484185876
// MI455X (gfx1250) — compile-verified
//
#include <hip/hip_runtime.h>

// =====================================================================
// ST-GCN generator for MI455X (gfx1250, wave32).
// All GEMM-like ops (1x1 convs, graph adjacency einsums, spatial pool
// matrices, 3-tap temporal convs) run through one generic batched,
// strided WMMA kernel built on V_WMMA_F32_16X16X4_F32 (full fp32
// precision, fp32 accumulate). The inner K-loop is branchless: all
// bounds guards are clamped addresses + v_cndmask selects (EXEC is
// never touched, as WMMA requires). Loads for a K-step are issued as
// one group, then 4 WMMAs consume them; latency is hidden by wave
// occupancy rather than intra-wave pipelining (which caused spills).
// =====================================================================

typedef __attribute__((ext_vector_type(2))) float v2f;
typedef __attribute__((ext_vector_type(8))) float v8f;

#define LEAKY_SLOPE 0.2f
#define INV_SQRT2 0.70710678118654752f

// ---------------------------------------------------------------------
// Generic WMMA f32 GEMM (template on A column stride so the common
// ACS==1 case emits contiguous/mergeable A loads):
//   for z in grid.z (batch):
//     D[z][m,n] = post( sum_{kb<kBatch} sum_k A[z,kb][m,k] * B[z,kb][k, n+colOff(kb)] )
//   A[z,kb][m,k] = A[z*aBatch + kb*aKStride + m*lda + k*ACS]
//   B[z,kb][k,c] = B[z*bBatch + kb*bKStride + k*ldb + c]   (c outside [0,bColMax) -> 0)
//   post: +bias[m], activation (0 none / 1 relu / 2 leaky 0.2),
//         optional residual: (v + add[z][m,n]) * scale, else v * scale.
//
// One wave computes a 16(M) x 64(N) strip = 4 D-tiles, reusing its A
// fragment 4x.  f32 fragment layouts (ISA 7.12.2):
//   A 16x4 : lanes 0-15 hold M=lane, K={0,1}; lanes 16-31 hold K={2,3}.
//   B 4x16 : mirrored, N across lanes, K rows split by lane half.
//   C/D    : VGPR r -> M = r + 8*(lane>=16), N = lane&15.
// ---------------------------------------------------------------------
template <int ACS>
__global__ void wmma_gemm_f32_t(
    const float* __restrict__ A, const float* __restrict__ B, float* __restrict__ D,
    int M, int N, int K,
    int lda, int ldb, int ldd,
    long aBatch, long bBatch, long dBatch,
    int kBatch, long aKStride, long bKStride,
    int bColOff0, int bColOffStep, int bColMax,
    const float* __restrict__ bias, int act,
    const float* __restrict__ add, long addBatch, float scale)
{
    const int lane = threadIdx.x & 31;
    const int wave = threadIdx.x >> 5;
    const int half = lane >> 4;   // K-half for A/B fragments
    const int l15  = lane & 15;

    const int mTiles  = (M + 15) >> 4;
    const int nTiles  = (N + 15) >> 4;
    const int nGroups = (nTiles + 3) >> 2;

    long tile = (long)blockIdx.x * (blockDim.x >> 5) + wave;
    if (tile >= (long)mTiles * nGroups) return;          // wave-uniform
    const int mt = (int)(tile / nGroups);
    const int ng = (int)(tile % nGroups);
    const int m0 = mt << 4;
    const int n0 = ng << 6;                              // base column of 4-tile strip
    const int z  = blockIdx.z;

    const float* Ab = A + (long)z * aBatch;
    const float* Bb = B + (long)z * bBatch;

    const v8f vzero = {0.f, 0.f, 0.f, 0.f, 0.f, 0.f, 0.f, 0.f};
    v8f acc[4];
#pragma unroll
    for (int j = 0; j < 4; ++j) acc[j] = vzero;

    // A row for this lane: clamp + select (branchless edge handling).
    const int  am  = m0 + l15;
    const bool aOK = (am < M);
    const int  amc = aOK ? am : (M - 1);
    const long aRowOff = (long)amc * lda + (long)(half << 1) * ACS;

    const int  kFull = K >> 2;   // unguarded 4-deep steps
    const int  kRem  = K & 3;
    const long aStep = 4L * ACS;
    const long bStep = 4L * ldb;

    for (int kb = 0; kb < kBatch; ++kb) {
        const int bOff = bColOff0 + kb * bColOffStep;

        // Per-lane streaming pointers (all address math hoisted).
        const float* ap   = Ab + (long)kb * aKStride + aRowOff;
        const float* bRow = Bb + (long)kb * bKStride + (long)(half << 1) * ldb;

        const float* bp0; const float* bp1; const float* bp2; const float* bp3;
        bool bok0, bok1, bok2, bok3;
        {
            int c0 = n0 + 0  + l15 + bOff;  bok0 = (c0 >= 0) && (c0 < bColMax);  bp0 = bRow + (bok0 ? c0 : 0);
            int c1 = n0 + 16 + l15 + bOff;  bok1 = (c1 >= 0) && (c1 < bColMax);  bp1 = bRow + (bok1 ? c1 : 0);
            int c2 = n0 + 32 + l15 + bOff;  bok2 = (c2 >= 0) && (c2 < bColMax);  bp2 = bRow + (bok2 ? c2 : 0);
            int c3 = n0 + 48 + l15 + bOff;  bok3 = (c3 >= 0) && (c3 < bColMax);  bp3 = bRow + (bok3 ? c3 : 0);
        }

        for (int s = 0; s < kFull; ++s) {
            // Issue the whole step's loads as one group (clause-able),
            // then consume with selects + 4 WMMAs into in-place accs.
            float ra0 = ap[0], ra1 = ap[ACS];
            float r00 = bp0[0], r01 = bp0[ldb];
            float r10 = bp1[0], r11 = bp1[ldb];
            float r20 = bp2[0], r21 = bp2[ldb];
            float r30 = bp3[0], r31 = bp3[ldb];
            ap += aStep; bp0 += bStep; bp1 += bStep; bp2 += bStep; bp3 += bStep;

            v2f a;
            a.x = aOK ? ra0 : 0.0f;
            a.y = aOK ? ra1 : 0.0f;
            {
                v2f b; b.x = bok0 ? r00 : 0.0f; b.y = bok0 ? r01 : 0.0f;
                acc[0] = __builtin_amdgcn_wmma_f32_16x16x4_f32(false, a, false, b, (short)0, acc[0], false, false);
            }
            {
                v2f b; b.x = bok1 ? r10 : 0.0f; b.y = bok1 ? r11 : 0.0f;
                acc[1] = __builtin_amdgcn_wmma_f32_16x16x4_f32(false, a, false, b, (short)0, acc[1], false, false);
            }
            {
                v2f b; b.x = bok2 ? r20 : 0.0f; b.y = bok2 ? r21 : 0.0f;
                acc[2] = __builtin_amdgcn_wmma_f32_16x16x4_f32(false, a, false, b, (short)0, acc[2], false, false);
            }
            {
                v2f b; b.x = bok3 ? r30 : 0.0f; b.y = bok3 ? r31 : 0.0f;
                acc[3] = __builtin_amdgcn_wmma_f32_16x16x4_f32(false, a, false, b, (short)0, acc[3], false, false);
            }
        }

        if (kRem) {
            // One guarded tail step: clamp K-rows, select-zero invalid halves.
            const int  ka   = (kFull << 2) + (half << 1);
            const bool k0ok = (ka < K);
            const bool k1ok = (ka + 1 < K);
            const int  r0   = k0ok ? ka : (K - 1);
            const int  r1   = k1ok ? (ka + 1) : (K - 1);

            const float* Ak0 = Ab + (long)kb * aKStride + (long)amc * lda;
            v2f a;
            {
                float t0 = Ak0[(long)r0 * ACS];
                float t1 = Ak0[(long)r1 * ACS];
                a.x = (aOK && k0ok) ? t0 : 0.0f;
                a.y = (aOK && k1ok) ? t1 : 0.0f;
            }
            const float* B0 = Bb + (long)kb * bKStride;
            const float* q0 = B0 + (long)r0 * ldb;
            const float* q1 = B0 + (long)r1 * ldb;
#pragma unroll
            for (int j = 0; j < 4; ++j) {
                int  bc  = n0 + (j << 4) + l15 + bOff;
                bool ok  = (bc >= 0) && (bc < bColMax);
                int  bcc = ok ? bc : 0;
                float u0 = q0[bcc];
                float u1 = q1[bcc];
                v2f b;
                b.x = (ok && k0ok) ? u0 : 0.0f;
                b.y = (ok && k1ok) ? u1 : 0.0f;
                acc[j] = __builtin_amdgcn_wmma_f32_16x16x4_f32(false, a, false, b, (short)0, acc[j], false, false);
            }
        }
    }

    float*       Dz = D + (long)z * dBatch;
    const float* Az = add ? (add + (long)z * addBatch) : (const float*)0;
#pragma unroll
    for (int j = 0; j < 4; ++j) {
        const int n = n0 + (j << 4) + l15;
        if (n < N) {
#pragma unroll
            for (int r = 0; r < 8; ++r) {
                const int m = m0 + r + (half << 3);
                if (m < M) {
                    float v = acc[j][r];
                    if (bias) v += bias[m];
                    if (act == 1)      v = fmaxf(v, 0.0f);
                    else if (act == 2) v = (v > 0.0f) ? v : LEAKY_SLOPE * v;
                    if (Az) v = (v + Az[(long)m * ldd + n]) * scale;
                    else    v *= scale;
                    Dz[(long)m * ldd + n] = v;
                }
            }
        }
    }
}

// ---------------------------------------------------------------------
// InstanceNorm over (T,V) per (n,c). mode 0: y = g[c]*nrm + b[c];
// mode 1 (AdaIN): y = (1+ada[n,c])*nrm + ada[n,C+c]. Optional leaky 0.2.
// ---------------------------------------------------------------------
__global__ void instnorm_kernel(const float* __restrict__ x, float* __restrict__ y,
                                int C, long TV,
                                const float* __restrict__ g, const float* __restrict__ b,
                                const float* __restrict__ ada, int mode, int leaky)
{
    const int c = blockIdx.x;
    const int n = blockIdx.y;
    const float* xp = x + ((long)n * C + c) * TV;
    float*       yp = y + ((long)n * C + c) * TV;

    float s = 0.f, s2 = 0.f;
    for (long i = threadIdx.x; i < TV; i += blockDim.x) {
        float v = xp[i]; s += v; s2 += v * v;
    }
    __shared__ float sh[256];
    __shared__ float sh2[256];
    sh[threadIdx.x]  = s;
    sh2[threadIdx.x] = s2;
    __syncthreads();
    for (int off = blockDim.x >> 1; off > 0; off >>= 1) {
        if ((int)threadIdx.x < off) {
            sh[threadIdx.x]  += sh[threadIdx.x + off];
            sh2[threadIdx.x] += sh2[threadIdx.x + off];
        }
        __syncthreads();
    }
    const float inv  = 1.0f / (float)TV;
    const float mean = sh[0] * inv;
    const float var  = sh2[0] * inv - mean * mean;
    const float rstd = rsqrtf(var + 1e-5f);

    float gamma, beta;
    if (mode == 0) { gamma = g[c];                            beta = b[c]; }
    else           { gamma = 1.0f + ada[(long)n * 2 * C + c]; beta = ada[(long)n * 2 * C + C + c]; }

    for (long i = threadIdx.x; i < TV; i += blockDim.x) {
        float v = (xp[i] - mean) * rstd * gamma + beta;
        if (leaky) v = (v > 0.0f) ? v : LEAKY_SLOPE * v;
        yp[i] = v;
    }
}

// AvgPool2d((2,1)) over T, optional relu.
__global__ void avgpool_t_kernel(const float* __restrict__ x, float* __restrict__ y,
                                 int C, int T, int V, int relu, int Nb)
{
    const int Th = T >> 1;
    const long total = (long)Nb * C * Th * V;
    long i = (long)blockIdx.x * blockDim.x + threadIdx.x;
    if (i >= total) return;
    int  v = (int)(i % V);  long r = i / V;
    int  t = (int)(r % Th); r /= Th;
    int  c = (int)(r % C);
    int  n = (int)(r / C);
    const float* xp = x + (((long)n * C + c) * T) * V;
    float val = 0.5f * (xp[(2 * t) * (long)V + v] + xp[(2 * t + 1) * (long)V + v]);
    if (relu) val = fmaxf(val, 0.0f);
    y[i] = val;
}

// Bilinear upsample x2 over T, align_corners=True, optional relu.
__global__ void unpool_t_kernel(const float* __restrict__ x, float* __restrict__ y,
                                int C, int T, int V, int relu, int Nb)
{
    const int T2 = 2 * T;
    const long total = (long)Nb * C * T2 * V;
    long i = (long)blockIdx.x * blockDim.x + threadIdx.x;
    if (i >= total) return;
    int  v  = (int)(i % V);   long r = i / V;
    int  t2 = (int)(r % T2);  r /= T2;
    int  c  = (int)(r % C);
    int  n  = (int)(r / C);
    const float scaleP = (float)(T - 1) / (float)(T2 - 1);
    float pos = (float)t2 * scaleP;
    int   i0  = (int)floorf(pos);
    int   i1  = min(i0 + 1, T - 1);
    float w   = pos - (float)i0;
    const float* xp = x + ((long)n * C + c) * (long)T * V;
    float val = xp[(long)i0 * V + v] * (1.0f - w) + xp[(long)i1 * V + v] * w;
    if (relu) val = fmaxf(val, 0.0f);
    y[i] = val;
}

// Style FC: out[n,o] = b[o] + dot(s[n,:sd], w[o,:sd])
__global__ void fc_kernel(const float* __restrict__ s, const float* __restrict__ w,
                          const float* __restrict__ b, float* __restrict__ out,
                          int sd, int od, int Nb)
{
    long i = (long)blockIdx.x * blockDim.x + threadIdx.x;
    if (i >= (long)Nb * od) return;
    int o = (int)(i % od);
    int n = (int)(i / od);
    const float* sp = s + (long)n * sd;
    const float* wp = w + (long)o * sd;
    float acc = b[o];
    for (int d = 0; d < sd; ++d) acc += sp[d] * wp[d];
    out[i] = acc;
}

// ---------------------------------------------------------------------
// Host-side orchestration
// ---------------------------------------------------------------------
static void gemm(hipStream_t st,
                 const float* A, const float* B, float* D,
                 int M, int N, int K, int lda, int aColStride, int ldb, int ldd,
                 long aBatch, long bBatch, long dBatch, int nb,
                 int kBatch, long aKStride, long bKStride,
                 int bColOff0, int bColOffStep, int bColMax,
                 const float* bias, int act, const float* add, long addBatch, float scale)
{
    const int mTiles  = (M + 15) / 16;
    const int nTiles  = (N + 15) / 16;
    const int nGroups = (nTiles + 3) / 4;
    const long tiles  = (long)mTiles * nGroups;
    dim3 grid((unsigned)((tiles + 3) / 4), 1, (unsigned)nb);
    if (aColStride == 3) {
        wmma_gemm_f32_t<3><<<grid, 128, 0, st>>>(A, B, D, M, N, K, lda, ldb, ldd,
                                                 aBatch, bBatch, dBatch, kBatch, aKStride, bKStride,
                                                 bColOff0, bColOffStep, bColMax,
                                                 bias, act, add, addBatch, scale);
    } else {
        wmma_gemm_f32_t<1><<<grid, 128, 0, st>>>(A, B, D, M, N, K, lda, ldb, ldd,
                                                 aBatch, bBatch, dBatch, kBatch, aKStride, bKStride,
                                                 bColOff0, bColOffStep, bColMax,
                                                 bias, act, add, addBatch, scale);
    }
}

struct EncParams {
    const float *c1_s_w, *c1_s_b, *c1_t_w, *c1_t_b;
    const float *c2_s_w, *c2_s_b, *c2_t_w, *c2_t_b;
    const float *n1_g, *n1_b, *n2_g, *n2_b, *proj_w;
};
struct DecParams {
    const float *ada1_w, *ada1_b, *ada2_w, *ada2_b;
    const float *c1_s_w, *c1_s_b, *c1_t_w, *c1_t_b;
    const float *c2_s_w, *c2_s_b, *c2_t_w, *c2_t_b, *proj_w;
};

// enc: in P (Ci,T,V) -> out P (Co,T/2,W)
static void enc_block(hipStream_t st, int Nb, int Ci, int Co, int T, int V, int W,
                      const float* Aadj, const float* Badj, const float* Mp,
                      const EncParams& p, float* P, float* S, float* B1, float* B2)
{
    const long TV  = (long)T * V;
    const int  Th  = T / 2;
    const long TW  = (long)T * W;
    const long ThW = (long)Th * W;

    // shortcut: S = avgpool_t(s_pool(conv1x1(P, proj), Mp))
    gemm(st, p.proj_w, P, B1, Co, (int)TV, Ci, Ci, 1, (int)TV, (int)TV,
         0, (long)Ci * TV, (long)Co * TV, Nb, 1, 0, 0, 0, 0, (int)TV,
         nullptr, 0, nullptr, 0, 1.0f);
    gemm(st, B1, Mp, B2, Co * T, W, V, V, 1, W, W,
         (long)Co * TV, 0, (long)Co * TW, Nb, 1, 0, 0, 0, 0, W,
         nullptr, 0, nullptr, 0, 1.0f);
    {
        long tot = (long)Nb * Co * Th * W;
        avgpool_t_kernel<<<dim3((unsigned)((tot + 255) / 256)), 256, 0, st>>>(B2, S, Co, T, W, 0, Nb);
    }
    // h = leaky(IN1(P))
    instnorm_kernel<<<dim3(Ci, Nb), 256, 0, st>>>(P, B1, Ci, TV, p.n1_g, p.n1_b, nullptr, 0, 1);
    // z1 = 1x1 to 3*Ci channels (+bias)
    gemm(st, p.c1_s_w, B1, B2, 3 * Ci, (int)TV, Ci, Ci, 1, (int)TV, (int)TV,
         0, (long)Ci * TV, 3L * Ci * TV, Nb, 1, 0, 0, 0, 0, (int)TV,
         p.c1_s_b, 0, nullptr, 0, 1.0f);
    // g1 = sum_k z1_k x Aadj_k
    gemm(st, B2, Aadj, B1, Ci * T, V, V, V, 1, V, V,
         3L * Ci * TV, 0, (long)Ci * TV, Nb, 3, (long)Ci * TV, (long)V * V, 0, 0, V,
         nullptr, 0, nullptr, 0, 1.0f);
    // s_pool
    gemm(st, B1, Mp, B2, Ci * T, W, V, V, 1, W, W,
         (long)Ci * TV, 0, (long)Ci * TW, Nb, 1, 0, 0, 0, 0, W,
         nullptr, 0, nullptr, 0, 1.0f);
    // relu(avgpool)
    {
        long tot = (long)Nb * Ci * Th * W;
        avgpool_t_kernel<<<dim3((unsigned)((tot + 255) / 256)), 256, 0, st>>>(B2, B1, Ci, T, W, 1, Nb);
    }
    // t_conv c1 (3-tap over T, zero-padded via B column shift)
    gemm(st, p.c1_t_w, B1, B2, Ci, (int)ThW, Ci, Ci * 3, 3, (int)ThW, (int)ThW,
         0, (long)Ci * ThW, (long)Ci * ThW, Nb, 3, 1, 0, -W, W, (int)ThW,
         p.c1_t_b, 0, nullptr, 0, 1.0f);
    // leaky(IN2)
    instnorm_kernel<<<dim3(Ci, Nb), 256, 0, st>>>(B2, B1, Ci, ThW, p.n2_g, p.n2_b, nullptr, 0, 1);
    // z2 = 1x1 to 3*Co channels (+bias)
    gemm(st, p.c2_s_w, B1, B2, 3 * Co, (int)ThW, Ci, Ci, 1, (int)ThW, (int)ThW,
         0, (long)Ci * ThW, 3L * Co * ThW, Nb, 1, 0, 0, 0, 0, (int)ThW,
         p.c2_s_b, 0, nullptr, 0, 1.0f);
    // relu(adjacency Badj on W-node graph)
    gemm(st, B2, Badj, B1, Co * Th, W, W, W, 1, W, W,
         3L * Co * ThW, 0, (long)Co * ThW, Nb, 3, (long)Co * ThW, (long)W * W, 0, 0, W,
         nullptr, 1, nullptr, 0, 1.0f);
    // (t_conv c2 + S) * 1/sqrt(2) -> P
    gemm(st, p.c2_t_w, B1, P, Co, (int)ThW, Co, Co * 3, 3, (int)ThW, (int)ThW,
         0, (long)Co * ThW, (long)Co * ThW, Nb, 3, 1, 0, -W, W, (int)ThW,
         p.c2_t_b, 0, S, (long)Co * ThW, INV_SQRT2);
}

// dec: in P (Ci,T,V) -> out P (Co,2T,W)
static void dec_block(hipStream_t st, int Nb, int Ci, int Co, int T, int V, int W, int sd,
                      const float* Aadj, const float* Badj, const float* Up, const float* sIn,
                      const DecParams& p, float* P, float* S, float* B1, float* B2, float* ADA)
{
    const long TV  = (long)T * V;
    const long TW  = (long)T * W;
    const int  T2  = 2 * T;
    const long T2W = (long)T2 * W;

    // shortcut: S = conv1x1(unpool(s_pool(P, Up)), proj)
    gemm(st, P, Up, B1, Ci * T, W, V, V, 1, W, W,
         (long)Ci * TV, 0, (long)Ci * TW, Nb, 1, 0, 0, 0, 0, W,
         nullptr, 0, nullptr, 0, 1.0f);
    {
        long tot = (long)Nb * Ci * T2 * W;
        unpool_t_kernel<<<dim3((unsigned)((tot + 255) / 256)), 256, 0, st>>>(B1, B2, Ci, T, W, 0, Nb);
    }
    gemm(st, p.proj_w, B2, S, Co, (int)T2W, Ci, Ci, 1, (int)T2W, (int)T2W,
         0, (long)Ci * T2W, (long)Co * T2W, Nb, 1, 0, 0, 0, 0, (int)T2W,
         nullptr, 0, nullptr, 0, 1.0f);

    // h = leaky(AdaIN1(P))
    {
        int od = 2 * Ci; long tot = (long)Nb * od;
        fc_kernel<<<dim3((unsigned)((tot + 255) / 256)), 256, 0, st>>>(sIn, p.ada1_w, p.ada1_b, ADA, sd, od, Nb);
    }
    instnorm_kernel<<<dim3(Ci, Nb), 256, 0, st>>>(P, B1, Ci, TV, nullptr, nullptr, ADA, 1, 1);

    // z1 = 1x1 Ci -> 3*Co (+bias)
    gemm(st, p.c1_s_w, B1, B2, 3 * Co, (int)TV, Ci, Ci, 1, (int)TV, (int)TV,
         0, (long)Ci * TV, 3L * Co * TV, Nb, 1, 0, 0, 0, 0, (int)TV,
         p.c1_s_b, 0, nullptr, 0, 1.0f);
    // adjacency Aadj (V-node graph)
    gemm(st, B2, Aadj, B1, Co * T, V, V, V, 1, V, V,
         3L * Co * TV, 0, (long)Co * TV, Nb, 3, (long)Co * TV, (long)V * V, 0, 0, V,
         nullptr, 0, nullptr, 0, 1.0f);
    // s_pool Up (V -> W)
    gemm(st, B1, Up, B2, Co * T, W, V, V, 1, W, W,
         (long)Co * TV, 0, (long)Co * TW, Nb, 1, 0, 0, 0, 0, W,
         nullptr, 0, nullptr, 0, 1.0f);
    // relu(unpool)
    {
        long tot = (long)Nb * Co * T2 * W;
        unpool_t_kernel<<<dim3((unsigned)((tot + 255) / 256)), 256, 0, st>>>(B2, B1, Co, T, W, 1, Nb);
    }
    // t_conv c1
    gemm(st, p.c1_t_w, B1, B2, Co, (int)T2W, Co, Co * 3, 3, (int)T2W, (int)T2W,
         0, (long)Co * T2W, (long)Co * T2W, Nb, 3, 1, 0, -W, W, (int)T2W,
         p.c1_t_b, 0, nullptr, 0, 1.0f);
    // leaky(AdaIN2)
    {
        int od = 2 * Co; long tot = (long)Nb * od;
        fc_kernel<<<dim3((unsigned)((tot + 255) / 256)), 256, 0, st>>>(sIn, p.ada2_w, p.ada2_b, ADA, sd, od, Nb);
    }
    instnorm_kernel<<<dim3(Co, Nb), 256, 0, st>>>(B2, B1, Co, T2W, nullptr, nullptr, ADA, 1, 1);
    // z2 = 1x1 Co -> 3*Co (+bias)
    gemm(st, p.c2_s_w, B1, B2, 3 * Co, (int)T2W, Co, Co, 1, (int)T2W, (int)T2W,
         0, (long)Co * T2W, 3L * Co * T2W, Nb, 1, 0, 0, 0, 0, (int)T2W,
         p.c2_s_b, 0, nullptr, 0, 1.0f);
    // relu(adjacency Badj on W-node graph)
    gemm(st, B2, Badj, B1, Co * T2, W, W, W, 1, W, W,
         3L * Co * T2W, 0, (long)Co * T2W, Nb, 3, (long)Co * T2W, (long)W * W, 0, 0, W,
         nullptr, 1, nullptr, 0, 1.0f);
    // (t_conv c2 + S) * 1/sqrt(2) -> P
    gemm(st, p.c2_t_w, B1, P, Co, (int)T2W, Co, Co * 3, 3, (int)T2W, (int)T2W,
         0, (long)Co * T2W, (long)Co * T2W, Nb, 3, 1, 0, -W, W, (int)T2W,
         p.c2_t_b, 0, S, (long)Co * T2W, INV_SQRT2);
}

// d_in layout: setup_inputs() insertion order for the top level, with the
// nested `params` dict flattened jax-pytree style (dict keys sorted
// alphabetically, recursively).
enum {
    IN_X = 0, IN_S, IN_AV1, IN_AV2, IN_AV3, IN_M12, IN_M23, IN_U32, IN_U21,
    D1_ADA1_B, D1_ADA1_W, D1_ADA2_B, D1_ADA2_W,
    D1_C1_S_B, D1_C1_S_W, D1_C1_T_B, D1_C1_T_W,
    D1_C2_S_B, D1_C2_S_W, D1_C2_T_B, D1_C2_T_W, D1_PROJ_W,
    D2_ADA1_B, D2_ADA1_W, D2_ADA2_B, D2_ADA2_W,
    D2_C1_S_B, D2_C1_S_W, D2_C1_T_B, D2_C1_T_W,
    D2_C2_S_B, D2_C2_S_W, D2_C2_T_B, D2_C2_T_W, D2_PROJ_W,
    E1_C1_S_B, E1_C1_S_W, E1_C1_T_B, E1_C1_T_W,
    E1_C2_S_B, E1_C2_S_W, E1_C2_T_B, E1_C2_T_W,
    E1_N1_B, E1_N1_G, E1_N2_B, E1_N2_G, E1_PROJ_W,
    E2_C1_S_B, E2_C1_S_W, E2_C1_T_B, E2_C1_T_W,
    E2_C2_S_B, E2_C2_S_W, E2_C2_T_B, E2_C2_T_W,
    E2_N1_B, E2_N1_G, E2_N2_B, E2_N2_G, E2_PROJ_W,
    FROM_XYZ_B, FROM_XYZ_W, TO_XYZ_B, TO_XYZ_W
};

extern "C" void kernel_launch(void* const* d_in, const int* in_sizes, int n_in,
                              void* d_out, int out_size, void* d_ws, size_t ws_size,
                              hipStream_t stream)
{
    (void)in_sizes; (void)n_in; (void)out_size; (void)ws_size;

    const int Nb = 16, T0 = 512, V1 = 28, V2 = 14, V3 = 7, SD = 64;
    auto in = [&](int i) { return (const float*)d_in[i]; };

    // Workspace arenas (floats): P 15M | S 15M | B1 30M | B2 45M | ADA small
    float* ws  = (float*)d_ws;
    float* P   = ws;
    float* S   = ws + 15000000L;
    float* B1  = ws + 30000000L;
    float* B2  = ws + 60000000L;
    float* ADA = ws + 105000000L;

    EncParams e1 = { in(E1_C1_S_W), in(E1_C1_S_B), in(E1_C1_T_W), in(E1_C1_T_B),
                     in(E1_C2_S_W), in(E1_C2_S_B), in(E1_C2_T_W), in(E1_C2_T_B),
                     in(E1_N1_G), in(E1_N1_B), in(E1_N2_G), in(E1_N2_B), in(E1_PROJ_W) };
    EncParams e2 = { in(E2_C1_S_W), in(E2_C1_S_B), in(E2_C1_T_W), in(E2_C1_T_B),
                     in(E2_C2_S_W), in(E2_C2_S_B), in(E2_C2_T_W), in(E2_C2_T_B),
                     in(E2_N1_G), in(E2_N1_B), in(E2_N2_G), in(E2_N2_B), in(E2_PROJ_W) };
    DecParams d1 = { in(D1_ADA1_W), in(D1_ADA1_B), in(D1_ADA2_W), in(D1_ADA2_B),
                     in(D1_C1_S_W), in(D1_C1_S_B), in(D1_C1_T_W), in(D1_C1_T_B),
                     in(D1_C2_S_W), in(D1_C2_S_B), in(D1_C2_T_W), in(D1_C2_T_B), in(D1_PROJ_W) };
    DecParams d2 = { in(D2_ADA1_W), in(D2_ADA1_B), in(D2_ADA2_W), in(D2_ADA2_B),
                     in(D2_C1_S_W), in(D2_C1_S_B), in(D2_C1_T_W), in(D2_C1_T_B),
                     in(D2_C2_S_W), in(D2_C2_S_B), in(D2_C2_T_W), in(D2_C2_T_B), in(D2_PROJ_W) };

    const long TV1 = (long)T0 * V1;   // 14336

    // from_xyz: (6 -> 64) over (T0, V1)
    gemm(stream, in(FROM_XYZ_W), in(IN_X), P, 64, (int)TV1, 6, 6, 1, (int)TV1, (int)TV1,
         0, 6L * TV1, 64L * TV1, Nb, 1, 0, 0, 0, 0, (int)TV1,
         in(FROM_XYZ_B), 0, nullptr, 0, 1.0f);

    // Encoder
    enc_block(stream, Nb,  64, 128, 512, V1, V2, in(IN_AV1), in(IN_AV2), in(IN_M12), e1, P, S, B1, B2);
    enc_block(stream, Nb, 128, 256, 256, V2, V3, in(IN_AV2), in(IN_AV3), in(IN_M23), e2, P, S, B1, B2);

    // Decoder
    dec_block(stream, Nb, 256, 128, 128, V3, V2, SD, in(IN_AV3), in(IN_AV2), in(IN_U32), in(IN_S), d1, P, S, B1, B2, ADA);
    dec_block(stream, Nb, 128,  64, 256, V2, V1, SD, in(IN_AV2), in(IN_AV1), in(IN_U21), in(IN_S), d2, P, S, B1, B2, ADA);

    // to_xyz: (64 -> 6) over (T0, V1) -> d_out
    gemm(stream, in(TO_XYZ_W), P, (float*)d_out, 6, (int)TV1, 64, 64, 1, (int)TV1, (int)TV1,
         0, 64L * TV1, 6L * TV1, Nb, 1, 0, 0, 0, 0, (int)TV1,
         in(TO_XYZ_B), 0, nullptr, 0, 1.0f);
}